// Model_20839181320906
// MI455X (gfx1250) — compile-verified
//
#include <hip/hip_runtime.h>
#include <hip/hip_bf16.h>
#include <cstdint>

typedef __attribute__((ext_vector_type(2))) float v2f;
typedef __attribute__((ext_vector_type(8))) float v8f;

#define MAX_SPIKE 1e5f

// ---------------------------------------------------------------------------
// Zero the 4 cost accumulators [wc1, l2_1, wc2, l2_2]
// ---------------------------------------------------------------------------
__global__ void init_acc_kernel(float* acc) {
    if (threadIdx.x < 4) acc[threadIdx.x] = 0.0f;
}

// ---------------------------------------------------------------------------
// Stable per-row argsort of (input ++ bias-spike 1.0), SZ=1024 bitonic.
// MODE 0: key = exp(2*(1-x)) (input layer); MODE 1: key = x (hidden layer).
// Composite (key, original-index) ordering reproduces jnp.argsort stability.
// ---------------------------------------------------------------------------
template <int MODE>
__global__ void sort_rows_kernel(const float* __restrict__ in, int n_in,
                                 float* __restrict__ xs, int* __restrict__ idx_out) {
    constexpr int SZ = 1024;
    __shared__ float sk[SZ];
    __shared__ int   si[SZ];
    const int row = blockIdx.x;
    const int tid = threadIdx.x;
    const int n   = n_in + 1;

    float key;
    if (tid < n_in) {
        float v = in[(size_t)row * n_in + tid];
        key = (MODE == 0) ? expf(2.0f * (1.0f - v)) : v;
    } else if (tid == n_in) {
        key = 1.0f;                       // bias spike time
    } else {
        key = 3.0e38f;                    // padding sorts last (> MAX_SPIKE)
    }
    sk[tid] = key;
    si[tid] = tid;
    __syncthreads();

    for (int k = 2; k <= SZ; k <<= 1) {
        for (int j = k >> 1; j > 0; j >>= 1) {
            int partner = tid ^ j;
            if (partner > tid) {
                bool up = ((tid & k) == 0);
                float a = sk[tid], b = sk[partner];
                int  ia = si[tid], ib = si[partner];
                bool a_gt_b = (a > b) || ((a == b) && (ia > ib)); // strict total order
                if (up ? a_gt_b : !a_gt_b) {
                    sk[tid] = b; sk[partner] = a;
                    si[tid] = ib; si[partner] = ia;
                }
            }
            __syncthreads();
        }
    }
    if (tid < n) {
        xs[(size_t)row * n + tid]      = sk[tid];
        idx_out[(size_t)row * n + tid] = si[tid];
    }
}

// ---------------------------------------------------------------------------
// SNN layer scan: one block per (row, 256-column tile). Walk sorted order,
// accumulate ws/wx, candidate out = wx / clip(ws-1, 1e-10, 1e10),
// valid = ws>1 && out < next spike time; min-reduce.
// Indices pre-scaled to row offsets in LDS -> 32-bit voffset addressing;
// 4-deep register prefetch pipeline + global_prefetch_b8 hides gather latency.
// ---------------------------------------------------------------------------
__global__ void snn_layer_kernel(const float* __restrict__ xs, const int* __restrict__ sidx,
                                 const float* __restrict__ W, float* __restrict__ out,
                                 int n, int outdim) {
    __shared__ float s_xs[1024];
    __shared__ int   s_off[1024];       // pre-scaled: idx * outdim (fits int32 easily)
    const int row = blockIdx.y;
    for (int i = threadIdx.x; i < n; i += blockDim.x) {
        s_xs[i]  = xs[(size_t)row * n + i];
        s_off[i] = sidx[(size_t)row * n + i] * outdim;
    }
    __syncthreads();

    const int col = blockIdx.x * blockDim.x + threadIdx.x;
    if (col >= outdim) return;

    float ws = 0.0f, wx = 0.0f, best = MAX_SPIKE;

    constexpr int PF = 4;
    float wbuf[PF];
#pragma unroll
    for (int u = 0; u < PF; ++u)
        wbuf[u] = W[(unsigned)(s_off[u] + col)];        // n >= 785 >> PF

    float xi = s_xs[0];                                  // rotate x through a register
    for (int base = 0; base < n; base += PF) {
        int pp = base + 2 * PF;                          // hint one pipeline depth ahead
        if (pp < n) __builtin_prefetch(&W[(unsigned)(s_off[pp] + col)], 0, 0);
#pragma unroll
        for (int u = 0; u < PF; ++u) {
            int i = base + u;
            if (i < n) {
                float w = wbuf[u];
                int pi = i + PF;
                wbuf[u] = (pi < n) ? W[(unsigned)(s_off[pi] + col)] : 0.0f;
                float xn = (i + 1 < n) ? s_xs[i + 1] : MAX_SPIKE;
                ws += w;
                wx += w * xi;
                float den = fminf(fmaxf(ws - 1.0f, 1e-10f), 1e10f);
                float o = wx / den;
                if ((ws > 1.0f) && (o < xn)) best = fminf(best, o);
                xi = xn;
            }
        }
    }
    out[(size_t)row * outdim + col] = best;
}

// ---------------------------------------------------------------------------
// w1 regularizer via WMMA: one 32-lane wave per 16 columns.
// A = ones(16x4); B = 4x16 chunk of W (and W^2) -> D row 0 accumulates the
// 16 column sums. 196 unguarded full chunks (running pointer) + 1 guarded
// tail chunk. Out-of-range columns clamp to cols-1; their D columns are
// never read. EXEC is all-ones at every WMMA (single full wave, converged).
// ---------------------------------------------------------------------------
__global__ void w_cost_wmma_kernel(const float* __restrict__ W, int rows, int cols,
                                   float* __restrict__ acc_wc, float* __restrict__ acc_l2) {
    const int lane  = threadIdx.x;      // block = 32
    const int nn    = lane & 15;        // output column within tile
    const int khalf = lane >> 4;        // which half of the K=4 slice
    int col = blockIdx.x * 16 + nn;
    const bool colok = (col < cols);
    if (!colok) col = cols - 1;         // clamp: duplicate sums land in unread D cols

#if __has_builtin(__builtin_amdgcn_wmma_f32_16x16x4_f32)
    v8f c  = {};
    v8f c2 = {};
    v2f a; a[0] = 1.0f; a[1] = 1.0f;    // all-ones A => D[m,n] = sum_k B[k,n]

    const int full = rows & ~3;         // 785 -> 784 (196 full chunks)
    const float* p = W + (size_t)(2 * khalf) * cols + col;
    for (int base = 0; base < full; base += 4) {
        float b0 = p[0];
        float b1 = p[cols];
        p += (size_t)4 * cols;
        v2f b;  b[0]  = b0;      b[1]  = b1;
        v2f bq; bq[0] = b0 * b0; bq[1] = b1 * b1;
        c  = __builtin_amdgcn_wmma_f32_16x16x4_f32(false, a, false, b,  (short)0, c,  false, false);
        c2 = __builtin_amdgcn_wmma_f32_16x16x4_f32(false, a, false, bq, (short)0, c2, false, false);
    }
    {   // tail chunk (rows full..full+3, only rows < `rows` contribute)
        int r0 = full + 2 * khalf;
        int r1 = r0 + 1;
        float b0 = (r0 < rows) ? W[(size_t)r0 * cols + col] : 0.0f;
        float b1 = (r1 < rows) ? W[(size_t)r1 * cols + col] : 0.0f;
        v2f b;  b[0]  = b0;      b[1]  = b1;
        v2f bq; bq[0] = b0 * b0; bq[1] = b1 * b1;
        c  = __builtin_amdgcn_wmma_f32_16x16x4_f32(false, a, false, b,  (short)0, c,  false, false);
        c2 = __builtin_amdgcn_wmma_f32_16x16x4_f32(false, a, false, bq, (short)0, c2, false, false);
    }
    if (lane < 16 && colok) {
        atomicAdd(acc_wc, fmaxf(0.0f, 1.0f - c[0]));   // lane<16, VGPR0: M=0, N=lane
        atomicAdd(acc_l2, c2[0]);
    }
#else
    if (colok) {
        float cs = 0.0f, l2 = 0.0f;
        for (int r = 0; r < rows; ++r) {
            float v = W[(size_t)r * cols + col];
            cs += v; l2 += v * v;
        }
        atomicAdd(acc_wc, fmaxf(0.0f, 1.0f - cs));
        atomicAdd(acc_l2, l2);
    }
#endif
}

// ---------------------------------------------------------------------------
// Small w2 regularizer (1001 x 10): plain reduction.
// ---------------------------------------------------------------------------
__global__ void w_cost_small_kernel(const float* __restrict__ W, int rows, int cols,
                                    float* __restrict__ acc_wc, float* __restrict__ acc_l2) {
    __shared__ float red[256];
    const int tid = threadIdx.x;
    float l2 = 0.0f;
    const int total = rows * cols;
    for (int e = tid; e < total; e += 256) { float v = W[e]; l2 += v * v; }
    red[tid] = l2;
    __syncthreads();
    for (int s = 128; s > 0; s >>= 1) {
        if (tid < s) red[tid] += red[tid + s];
        __syncthreads();
    }
    if (tid == 0) atomicAdd(acc_l2, red[0]);
    if (tid < cols) {
        float cs = 0.0f;
        for (int r = 0; r < rows; ++r) cs += W[(size_t)r * cols + tid];
        atomicAdd(acc_wc, fmaxf(0.0f, 1.0f - cs));
    }
}

// ---------------------------------------------------------------------------
// Loss + accuracy + total cost. One block of 64 threads (one per sample).
// ---------------------------------------------------------------------------
__global__ void finalize_kernel(const float* __restrict__ out2, const int* __restrict__ label,
                                const float* __restrict__ acc, float* __restrict__ d_out,
                                int B, int C) {
    __shared__ float s_loss[64];
    __shared__ float s_corr[64];
    const int b = threadIdx.x;
    float lossb = 0.0f, corrb = 0.0f;
    if (b < B) {
        int lbl = label[b];
        float z2 = 0.0f, o_true = 0.0f;
        float bestneg = -3.4e38f; int bestj = 0;
        for (int j = 0; j < C; ++j) {
            float o = out2[(size_t)b * C + j];
            z2 += expf(-o);
            if (-o > bestneg) { bestneg = -o; bestj = j; }   // first-max tiebreak
            if (j == lbl) o_true = o;
        }
        float z1 = expf(-o_true);
        z2 = fminf(fmaxf(z2, 1e-10f), 1e10f);
        float ratio = fminf(fmaxf(z1 / z2, 1e-10f), 1.0f);
        lossb = -logf(ratio);
        corrb = (bestj == lbl) ? 1.0f : 0.0f;
    }
    s_loss[b] = lossb; s_corr[b] = corrb;
    __syncthreads();
    for (int s = 32; s > 0; s >>= 1) {
        if (b < s) { s_loss[b] += s_loss[b + s]; s_corr[b] += s_corr[b + s]; }
        __syncthreads();
    }
    if (b == 0) {
        float loss = s_loss[0] / (float)B;
        float corr = s_corr[0] / (float)B;
        float cost = loss + 100.0f * (acc[0] + acc[2]) + 0.01f * (acc[1] + acc[3]);
        d_out[0] = cost;
        d_out[1] = corr;
    }
}

// ---------------------------------------------------------------------------
extern "C" void kernel_launch(void* const* d_in, const int* in_sizes, int n_in_args,
                              void* d_out, int out_size, void* d_ws, size_t ws_size,
                              hipStream_t stream) {
    (void)in_sizes; (void)n_in_args; (void)out_size; (void)ws_size;

    const int B = 64, IN = 784, HID = 1000, OUT = 10;
    const int N1 = IN + 1;    // 785
    const int N2 = HID + 1;   // 1001

    const float* image = (const float*)d_in[0];
    const int*   label = (const int*)d_in[1];
    const float* w1    = (const float*)d_in[2];
    const float* w2    = (const float*)d_in[3];
    float*       outp  = (float*)d_out;

    // workspace carve-up (all float/int32, 4-byte aligned)
    float* p    = (float*)d_ws;
    float* xs1  = p;                   p += (size_t)B * N1;
    int*   idx1 = (int*)p;             p += (size_t)B * N1;
    float* h    = p;                   p += (size_t)B * HID;
    float* xs2  = p;                   p += (size_t)B * N2;
    int*   idx2 = (int*)p;             p += (size_t)B * N2;
    float* out2 = p;                   p += (size_t)B * OUT;
    float* acc  = p;                   /* 4 floats: wc1, l2_1, wc2, l2_2 */

    init_acc_kernel<<<1, 32, 0, stream>>>(acc);

    // Layer 1: sort exp-transformed spikes, then scan.
    sort_rows_kernel<0><<<B, 1024, 0, stream>>>(image, IN, xs1, idx1);
    snn_layer_kernel<<<dim3((HID + 255) / 256, B), 256, 0, stream>>>(xs1, idx1, w1, h, N1, HID);

    // Layer 2.
    sort_rows_kernel<1><<<B, 1024, 0, stream>>>(h, HID, xs2, idx2);
    snn_layer_kernel<<<dim3(1, B), 256, 0, stream>>>(xs2, idx2, w2, out2, N2, OUT);

    // Regularizers: WMMA column-sum reduction on w1, scalar on tiny w2.
    w_cost_wmma_kernel<<<(HID + 15) / 16, 32, 0, stream>>>(w1, N1, HID, &acc[0], &acc[1]);
    w_cost_small_kernel<<<1, 256, 0, stream>>>(w2, N2, OUT, &acc[2], &acc[3]);

    // Loss / accuracy / total cost.
    finalize_kernel<<<1, 64, 0, stream>>>(out2, label, acc, outp, B, OUT);
}